// CustomRotatedRoIAlign_64819646431447
// MI455X (gfx1250) — compile-verified
//
#include <hip/hip_runtime.h>

// CDNA5 / gfx1250 wave32 WMMA types
typedef __attribute__((ext_vector_type(2))) float v2f;
typedef __attribute__((ext_vector_type(8))) float v8f;

#define BATCH 2
#define CHN   256
#define FH    200
#define FW    304
#define NBOX  512
#define OH    7
#define OW    7
#define NPTS  49   // OH*OW

__device__ __forceinline__ int iclamp(int v, int lo, int hi) {
    return v < lo ? lo : (v > hi ? hi : v);
}

// One wave = one (box, 16-channel group). 8 waves per block.
__global__ __launch_bounds__(256) void rroi_align_wmma_kernel(
    const float* __restrict__ fm,     // (B, C, H, W)
    const float* __restrict__ boxes,  // (B, N, 5)
    float* __restrict__ out)          // (B*N, C, 7, 7)
{
    __shared__ float lds[8][16 * NPTS];   // per-wave 16ch x 49pt tile (25 KB/block)

    const int lane  = threadIdx.x & 31;
    const int wslot = threadIdx.x >> 5;
    const int wid   = blockIdx.x * 8 + wslot;   // 0 .. 16383
    const int bn    = wid >> 4;                 // box 0..1023
    const int cg    = wid & 15;                 // channel group 0..15
    const int b     = bn / NBOX;

    // Box -> affine grid coefficients (redundant per lane; trivial VALU cost)
    const float cx  = boxes[bn * 5 + 0];
    const float cy  = boxes[bn * 5 + 1];
    const float bw  = boxes[bn * 5 + 2];
    const float bh  = boxes[bn * 5 + 3];
    const float ang = boxes[bn * 5 + 4];
    const float rad = ang * -0.017453292519943295f;
    const float sn = sinf(rad), cs = cosf(rad);
    const float a00 =  bw * (1.0f / FW) * cs;
    const float a01 = -bh * (1.0f / FH) * sn;
    const float a02 =  2.0f * cx * (1.0f / FW) - 1.0f;
    const float a10 =  bw * (1.0f / FW) * sn;
    const float a11 =  bh * (1.0f / FH) * cs;
    const float a12 =  2.0f * cy * (1.0f / FH) - 1.0f;

    const int hi = lane >> 4;        // 0: corners k0,k1 (row y0)   1: k2,k3 (row y1)
    const int cl = lane & 15;        // channel within group -> WMMA N column
    const int c  = cg * 16 + cl;
    const float* fmc = fm + (size_t)(b * CHN + c) * (FH * FW);

    for (int p = 0; p < NPTS; ++p) {
        const int oy = p / OW, ox = p - oy * OW;
        const float xs = (2.0f * ox + 1.0f) * (1.0f / OW) - 1.0f;
        const float ys = (2.0f * oy + 1.0f) * (1.0f / OH) - 1.0f;
        const float gx = a00 * xs + a01 * ys + a02;
        const float gy = a10 * xs + a11 * ys + a12;
        const float ix = ((gx + 1.0f) * FW - 1.0f) * 0.5f;
        const float iy = ((gy + 1.0f) * FH - 1.0f) * 0.5f;

        const float x0f = floorf(ix), y0f = floorf(iy);
        const float wx = ix - x0f,   wy = iy - y0f;
        const float x1f = x0f + 1.0f, y1f = y0f + 1.0f;

        // validity folded into the weights (matches reference vals*(wgt*valid))
        const float vx0 = (x0f >= 0.0f && x0f <= (float)(FW - 1)) ? 1.0f : 0.0f;
        const float vx1 = (x1f >= 0.0f && x1f <= (float)(FW - 1)) ? 1.0f : 0.0f;
        const float vy0 = (y0f >= 0.0f && y0f <= (float)(FH - 1)) ? 1.0f : 0.0f;
        const float vy1 = (y1f >= 0.0f && y1f <= (float)(FH - 1)) ? 1.0f : 0.0f;
        const float w00 = (1.0f - wy) * (1.0f - wx) * vy0 * vx0;  // k0 (y0,x0)
        const float w01 = (1.0f - wy) * wx          * vy0 * vx1;  // k1 (y0,x1)
        const float w10 = wy          * (1.0f - wx) * vy1 * vx0;  // k2 (y1,x0)
        const float w11 = wy          * wx          * vy1 * vx1;  // k3 (y1,x1)

        const int xi0 = iclamp((int)x0f, 0, FW - 1);
        const int xi1 = iclamp((int)x1f, 0, FW - 1);
        const int yi0 = iclamp((int)y0f, 0, FH - 1);
        const int yi1 = iclamp((int)y1f, 0, FH - 1);

        // A 16x4 f32: VGPR0 = K(2*hi), VGPR1 = K(1+2*hi); rows all identical
        v2f A;
        A[0] = hi ? w10 : w00;
        A[1] = hi ? w11 : w01;

        // B 4x16 f32: VGPR0 = corner K(2*hi), VGPR1 = K(1+2*hi), col = channel
        const float* rowp = fmc + (size_t)(hi ? yi1 : yi0) * FW;
        v2f Bv;
        Bv[0] = rowp[xi0];
        Bv[1] = rowp[xi1];

        v8f acc = {0.f, 0.f, 0.f, 0.f, 0.f, 0.f, 0.f, 0.f};
        // K=4 == the 4 bilinear corners; full f32, no precision loss
        acc = __builtin_amdgcn_wmma_f32_16x16x4_f32(
                  false, A, false, Bv, (short)0, acc, false, false);

        // every D row is identical; row 0 cols live in acc[0], lanes 0..15
        if (lane < 16) lds[wslot][cl * NPTS + p] = acc[0];
    }

    __syncthreads();  // uniform across all 8 waves

    // LDS tile is laid out exactly linearly with the output slice:
    // out[bn, cg*16 + cl, p] at offset cl*49 + p  -> fully coalesced stores
    float* obase = out + (size_t)bn * (CHN * NPTS) + (size_t)cg * (16 * NPTS);
    for (int i = lane; i < 16 * NPTS; i += 32)
        obase[i] = lds[wslot][i];
}

extern "C" void kernel_launch(void* const* d_in, const int* in_sizes, int n_in,
                              void* d_out, int out_size, void* d_ws, size_t ws_size,
                              hipStream_t stream) {
    const float* fm    = (const float*)d_in[0];
    const float* boxes = (const float*)d_in[1];
    float* out         = (float*)d_out;

    // 1024 boxes * 16 channel-groups = 16384 waves, 8 waves (256 thr) per block
    const int blocks = (BATCH * NBOX * 16) / 8;   // 2048
    rroi_align_wmma_kernel<<<blocks, 256, 0, stream>>>(fm, boxes, out);
}